// CriterionIFV_27264452395077
// MI455X (gfx1250) — compile-verified
//
#include <hip/hip_runtime.h>
#include <hip/hip_bf16.h>
#include <math.h>

// ---------------- problem constants (from reference setup_inputs) ----------
#define BB    8          // batch
#define CC    512        // channels
#define HH    128        // feature H
#define WWF   128        // feature W
#define HWP   (HH*WWF)   // 16384 pixels per batch
#define THW   512        // target H=W
#define NCLS  19
#define KPAD  32         // classes padded to 2 WMMA N-tiles
#define BINS  21         // per-thread LDS bin stride (odd -> conflict-free)

typedef __attribute__((ext_vector_type(2))) float v2f;
typedef __attribute__((ext_vector_type(8))) float v8f;

// ---------------- workspace layout (bytes) ---------------------------------
#define WS_LAB     0u                                  // BB*HWP ints      = 524288
#define WS_CNTS    524288u                             // BB*NCLS ints     (pad 1024)
#define WS_SUMS_S  525312u                             // BB*CC*NCLS f32   = 311296
#define WS_SUMS_T  836608u
#define WS_MEAN_S  1147904u                            // BB*CC*KPAD f32   = 524288
#define WS_MEAN_T  1672192u
#define WS_CN_S    2196480u                            // BB*KPAD f32 (pad 1024)
#define WS_CN_T    2197504u

// ---------------- K0: zero a float region ----------------------------------
__global__ void zero_kernel(float* __restrict__ p, int n) {
    int i = blockIdx.x * blockDim.x + threadIdx.x;
    if (i < n) p[i] = 0.0f;
}

// ---------------- K1: nearest-downsample labels + class counts -------------
__global__ void labels_kernel(const int* __restrict__ target,
                              int* __restrict__ lab, int* __restrict__ cnts) {
    int t = blockIdx.x * blockDim.x + threadIdx.x;
    if (t >= BB * HWP) return;
    int b = t / HWP, p = t % HWP;
    int h = p / WWF, w = p % WWF;
    // floor(i * 512/128) = 4i  -> stride-4 nearest pick
    int k = target[(size_t)b * THW * THW + (size_t)(h * 4) * THW + (w * 4)];
    lab[t] = k;
    if (k >= 0 && k < NCLS) atomicAdd(&cnts[b * NCLS + k], 1);
}

// ---------------- K2: per-(b,c) class sums (segmented reduction) -----------
// float4 loads (global_load_b128) + per-thread LDS histogram bins: ~4
// instructions per element instead of 38 VALU for a 19-way select tree.
__global__ __launch_bounds__(256) void sums_kernel(
    const float* __restrict__ featS, const float* __restrict__ featT,
    const int* __restrict__ lab,
    float* __restrict__ sumsS, float* __restrict__ sumsT) {
    __shared__ float bins[256 * BINS];   // 21504 B
    int blk = blockIdx.x;
    int sel = (blk >= BB * CC) ? 1 : 0;
    if (sel) blk -= BB * CC;
    const int b = blk / CC, c = blk % CC;
    const float* f = (sel ? featT : featS) + ((size_t)b * CC + c) * HWP;
    float* sums = sel ? sumsT : sumsS;
    const int* lb = lab + (size_t)b * HWP;
    const int tid = threadIdx.x;

    float* myb = &bins[tid * BINS];
#pragma unroll
    for (int j = 0; j < BINS; ++j) myb[j] = 0.0f;   // private bins, no sync needed

    const float4* f4 = (const float4*)f;
    const int4*   l4 = (const int4*)lb;
    for (int p = tid; p < HWP / 4; p += 256) {
        float4 v = f4[p];
        int4   k = l4[p];
        __builtin_prefetch(&f4[p + 1024], 0, 0);    // gfx1250 global_prefetch_b8
        myb[k.x] += v.x;                            // ds RMW, in-order per wave
        myb[k.y] += v.y;
        myb[k.z] += v.z;
        myb[k.w] += v.w;
    }
    __syncthreads();
    if (tid < NCLS) {
        float s = 0.0f;
        for (int t = 0; t < 256; ++t) s += bins[t * BINS + tid];
        sums[((size_t)b * CC + c) * NCLS + tid] = s;
    }
}

// ---------------- K3: means = sums/(cnt+1e-6), padded to KPAD --------------
__global__ void means_kernel(const float* __restrict__ sums,
                             const int* __restrict__ cnts,
                             float* __restrict__ means) {
    int t = blockIdx.x * blockDim.x + threadIdx.x;  // over BB*CC*KPAD
    if (t >= BB * CC * KPAD) return;
    int n  = t % KPAD;
    int bc = t / KPAD;
    int b  = bc / CC;
    float v = 0.0f;
    if (n < NCLS) v = sums[(size_t)bc * NCLS + n] / ((float)cnts[b * NCLS + n] + 1e-6f);
    means[(size_t)bc * KPAD + n] = v;
}

// ---------------- K3b: center norms per (b,class) --------------------------
__global__ void cnorm_kernel(const float* __restrict__ means,
                             float* __restrict__ cnorm) {
    const int bn = blockIdx.x;           // 0 .. BB*KPAD-1
    const int b = bn / KPAD, n = bn % KPAD;
    const int lane = threadIdx.x;        // 32 threads = 1 wave
    float s = 0.0f;
    for (int c = lane; c < CC; c += 32) {
        float v = means[((size_t)b * CC + c) * KPAD + n];
        s += v * v;
    }
    for (int off = 16; off >= 1; off >>= 1) s += __shfl_xor(s, off, 32);
    if (lane == 0) cnorm[bn] = sqrtf(s);
}

// ---------------- K4: WMMA cosine similarity + MSE loss --------------------
// One wave = one 16-pixel tile.  G[p, k] = sum_c feat[c,p]*means[c,k] via
// V_WMMA_F32_16X16X4_F32, K-loop over 512 channels in chunks of 4,
// two class tiles (0..15, 16..31).  ||feat||^2 folded into the A loads.
__global__ __launch_bounds__(256) void cosloss_kernel(
    const float* __restrict__ featS, const float* __restrict__ featT,
    const float* __restrict__ meansS, const float* __restrict__ meansT,
    const float* __restrict__ cnS, const float* __restrict__ cnT,
    const int* __restrict__ lab, float* __restrict__ out) {
    __shared__ float gbuf[8][16][KPAD];   // per-wave D-tile staging (16 KB)

    const int tid  = threadIdx.x;
    const int wave = tid >> 5;
    const int lane = tid & 31;
    const int half = lane >> 4;           // 0: K=0,1 rows; 1: K=2,3 rows
    const int l15  = lane & 15;

    const int tileId = blockIdx.x * 8 + wave;     // 8192 tiles total
    const int b  = tileId / (HWP / 16);
    const int p0 = (tileId % (HWP / 16)) * 16;

    const float* fS = featS  + (size_t)b * CC * HWP;
    const float* fT = featT  + (size_t)b * CC * HWP;
    const float* mS = meansS + (size_t)b * CC * KPAD;
    const float* mT = meansT + (size_t)b * CC * KPAD;

    v8f accS0 = {}, accS1 = {}, accT0 = {}, accT1 = {};
    float nrmS = 0.0f, nrmT = 0.0f;

    for (int c0 = 0; c0 < CC; c0 += 4) {
        const int    ccn = c0 + 2 * half;
        const size_t fo  = (size_t)ccn * HWP + p0 + l15;
        v2f a, bt0, bt1;
        // ---- student ----
        a.x = fS[fo]; a.y = fS[fo + HWP];                 // A[m=pixel, k=chan]
        bt0.x = mS[ccn * KPAD + l15];       bt0.y = mS[(ccn + 1) * KPAD + l15];
        bt1.x = mS[ccn * KPAD + 16 + l15];  bt1.y = mS[(ccn + 1) * KPAD + 16 + l15];
        nrmS += a.x * a.x + a.y * a.y;
        accS0 = __builtin_amdgcn_wmma_f32_16x16x4_f32(false, a, false, bt0, (short)0, accS0, false, false);
        accS1 = __builtin_amdgcn_wmma_f32_16x16x4_f32(false, a, false, bt1, (short)0, accS1, false, false);
        // ---- teacher ----
        a.x = fT[fo]; a.y = fT[fo + HWP];
        bt0.x = mT[ccn * KPAD + l15];       bt0.y = mT[(ccn + 1) * KPAD + l15];
        bt1.x = mT[ccn * KPAD + 16 + l15];  bt1.y = mT[(ccn + 1) * KPAD + 16 + l15];
        nrmT += a.x * a.x + a.y * a.y;
        accT0 = __builtin_amdgcn_wmma_f32_16x16x4_f32(false, a, false, bt0, (short)0, accT0, false, false);
        accT1 = __builtin_amdgcn_wmma_f32_16x16x4_f32(false, a, false, bt1, (short)0, accT1, false, false);
    }

    // lanes l and l+16 hold complementary channel halves of the same pixel
    nrmS += __shfl_xor(nrmS, 16, 32);
    nrmT += __shfl_xor(nrmT, 16, 32);

    const int m = l15;                                    // pixel within tile
    int k = lab[(size_t)b * HWP + p0 + m];
    if (k < 0) k = 0;
    if (k > NCLS - 1) k = NCLS - 1;

    // ---- stage + gather student G ----
#pragma unroll
    for (int r = 0; r < 8; ++r) gbuf[wave][r + 8 * half][l15]      = accS0[r];
#pragma unroll
    for (int r = 0; r < 8; ++r) gbuf[wave][r + 8 * half][16 + l15] = accS1[r];
    __syncthreads();
    const float gS = gbuf[wave][m][k];
    __syncthreads();
    // ---- stage + gather teacher G ----
#pragma unroll
    for (int r = 0; r < 8; ++r) gbuf[wave][r + 8 * half][l15]      = accT0[r];
#pragma unroll
    for (int r = 0; r < 8; ++r) gbuf[wave][r + 8 * half][16 + l15] = accT1[r];
    __syncthreads();
    const float gT = gbuf[wave][m][k];

    const float cS = cnS[b * KPAD + k];
    const float cT = cnT[b * KPAD + k];
    const float pS = gS / fmaxf(sqrtf(nrmS) * cS, 1e-8f);
    const float pT = gT / fmaxf(sqrtf(nrmT) * cT, 1e-8f);
    const float d  = pS - pT;

    float contrib = (half == 0) ? d * d : 0.0f;           // lanes 0..15 only
    for (int off = 16; off >= 1; off >>= 1) contrib += __shfl_down(contrib, off, 32);
    if (lane == 0) atomicAdd(out, contrib * (1.0f / (float)(BB * HWP)));
}

// ---------------- host launcher --------------------------------------------
extern "C" void kernel_launch(void* const* d_in, const int* in_sizes, int n_in,
                              void* d_out, int out_size, void* d_ws, size_t ws_size,
                              hipStream_t stream) {
    (void)in_sizes; (void)n_in; (void)out_size; (void)ws_size;
    const float* featS  = (const float*)d_in[0];
    const float* featT  = (const float*)d_in[1];
    const int*   target = (const int*)d_in[2];
    float*       out    = (float*)d_out;

    char* ws = (char*)d_ws;
    int*   lab    = (int*)  (ws + WS_LAB);
    int*   cnts   = (int*)  (ws + WS_CNTS);
    float* sumsS  = (float*)(ws + WS_SUMS_S);
    float* sumsT  = (float*)(ws + WS_SUMS_T);
    float* meansS = (float*)(ws + WS_MEAN_S);
    float* meansT = (float*)(ws + WS_MEAN_T);
    float* cnS    = (float*)(ws + WS_CN_S);
    float* cnT    = (float*)(ws + WS_CN_T);

    // K0: zero cnts..sumsT (contiguous) and the scalar output
    {
        const int nz = (int)((WS_MEAN_S - WS_CNTS) / 4);
        zero_kernel<<<(nz + 255) / 256, 256, 0, stream>>>((float*)(ws + WS_CNTS), nz);
        zero_kernel<<<1, 32, 0, stream>>>(out, 1);
    }
    // K1: labels + counts
    labels_kernel<<<(BB * HWP + 255) / 256, 256, 0, stream>>>(target, lab, cnts);
    // K2: class sums for both tensors (one block per (tensor,b,c))
    sums_kernel<<<2 * BB * CC, 256, 0, stream>>>(featS, featT, lab, sumsS, sumsT);
    // K3: means (padded)
    means_kernel<<<(BB * CC * KPAD + 255) / 256, 256, 0, stream>>>(sumsS, cnts, meansS);
    means_kernel<<<(BB * CC * KPAD + 255) / 256, 256, 0, stream>>>(sumsT, cnts, meansT);
    // K3b: center norms
    cnorm_kernel<<<BB * KPAD, 32, 0, stream>>>(meansS, cnS);
    cnorm_kernel<<<BB * KPAD, 32, 0, stream>>>(meansT, cnT);
    // K4: WMMA cosine-similarity MSE
    cosloss_kernel<<<(BB * HWP / 16) / 8, 256, 0, stream>>>(
        featS, featT, meansS, meansT, cnS, cnT, lab, out);
}